// SAGE_special_37194416783909
// MI455X (gfx1250) — compile-verified
//
#include <hip/hip_runtime.h>
#include <math.h>

// ---------------------------------------------------------------------------
// GraphSAGE (2-layer, mean aggregation) for MI455X / gfx1250 (wave32).
// Transform-then-aggregate: h = ELU( agg(X@Wl)/deg + X@Wr + b ).
// GEMMs use native fp32 WMMA (V_WMMA_F32_16X16X4_F32) => exact f32 math.
// Layer-1 weights are pre-padded to 48 cols so the GEMM is guard-free.
// ---------------------------------------------------------------------------

typedef float v2f __attribute__((ext_vector_type(2)));
typedef float v8f __attribute__((ext_vector_type(8)));

#define K_DIM 128   // both layers have K = 128

// One wave32 computes a 16x16 tile of BOTH  T = X@Wl  and  R = X@Wr + b.
// A (16x4 f32): lane 0-15 -> M=lane, K=0,1 ; lane 16-31 -> M=lane-16, K=2,3
// B (4x16 f32): v0/v1 = rows {0,1} in lanes 0-15, rows {2,3} in lanes 16-31
// C/D (16x16) : vgpr r, lanes 0-15 -> M=r, N=lane ; lanes 16-31 -> M=r+8
// CPAD is the (padded) column count of Wl/Wr/bias/T/R; no bounds checks.
template <int CPAD>
__global__ void __launch_bounds__(256)
gemm_dual_wmma(const float* __restrict__ X,    // n x 128
               const float* __restrict__ Wl,   // 128 x CPAD (row major)
               const float* __restrict__ Wr,   // 128 x CPAD
               const float* __restrict__ bias, // CPAD
               float* __restrict__ T,          // n x CPAD
               float* __restrict__ R)          // n x CPAD
{
  const int lane = threadIdx.x & 31;
  const int wave = threadIdx.x >> 5;       // col tile within block
  const int m0   = blockIdx.x << 4;        // 16-row tile
  const int n0   = wave << 4;              // 16-col tile
  const int rm   = lane & 15;
  const int half = lane >> 4;
  const int col  = n0 + rm;

  // Hoisted per-lane bases: every load below is base + compile-time offset.
  const float* xbase  = X  + (size_t)(m0 + rm) * K_DIM + (half << 1);
  const float* wlbase = Wl + (size_t)(half << 1) * CPAD + col;
  const float* wrbase = Wr + (size_t)(half << 1) * CPAD + col;

  v8f accL = {0.f,0.f,0.f,0.f,0.f,0.f,0.f,0.f};
  v8f accR = {0.f,0.f,0.f,0.f,0.f,0.f,0.f,0.f};

#pragma unroll
  for (int kb = 0; kb < K_DIM; kb += 4) {
    v2f a  = { xbase[kb],          xbase[kb + 1] };
    v2f bl = { wlbase[kb * CPAD],  wlbase[kb * CPAD + CPAD] };
    v2f br = { wrbase[kb * CPAD],  wrbase[kb * CPAD + CPAD] };
    accL = __builtin_amdgcn_wmma_f32_16x16x4_f32(
        false, a, false, bl, (short)0, accL, false, false);
    accR = __builtin_amdgcn_wmma_f32_16x16x4_f32(
        false, a, false, br, (short)0, accR, false, false);
  }

  const float bv = bias[col];
#pragma unroll
  for (int r = 0; r < 8; ++r) {
    const int row = m0 + r + (half << 3);
    T[(size_t)row * CPAD + col] = accL[r];
    R[(size_t)row * CPAD + col] = accR[r] + bv;
  }
}

// Zero-pad layer-1 params: W (128x40) -> Wp (128x48), b (40) -> bp (48).
__global__ void __launch_bounds__(256)
pad_l1_params(const float* __restrict__ Wl, const float* __restrict__ Wr,
              const float* __restrict__ b,
              float* __restrict__ Wlp, float* __restrict__ Wrp,
              float* __restrict__ bp)
{
  int t = blockIdx.x * blockDim.x + threadIdx.x;
  if (t >= 128 * 48) return;
  const int k = t / 48;
  const int c = t % 48;
  float vl = 0.0f, vr = 0.0f;
  if (c < 40) { vl = Wl[k * 40 + c]; vr = Wr[k * 40 + c]; }
  Wlp[t] = vl;
  Wrp[t] = vr;
  if (t < 48) bp[t] = (t < 40) ? b[t] : 0.0f;
}

// deg[i] = #incoming edges (as float)
__global__ void __launch_bounds__(256)
deg_kernel(const int* __restrict__ dst, float* __restrict__ deg, int E)
{
  int e = blockIdx.x * blockDim.x + threadIdx.x;
  if (e < E) atomicAdd(deg + dst[e], 1.0f);
}

// agg[dst] += T[src]   (float4 chunks; agg buffer lives in L2 at this size)
template <int CHUNKS, int STRIDE>
__global__ void __launch_bounds__(256)
scatter_add(const float* __restrict__ T, const int* __restrict__ src,
            const int* __restrict__ dst, float* __restrict__ agg, int E)
{
  long long t = (long long)blockIdx.x * blockDim.x + threadIdx.x;
  if (t >= (long long)E * CHUNKS) return;
  const int e = (int)(t / CHUNKS);
  const int c = (int)(t % CHUNKS);
  const int s = src[e];
  const int d = dst[e];
  const float4 v = *(const float4*)(T + (size_t)s * STRIDE + c * 4);
  float* o = agg + (size_t)d * STRIDE + c * 4;
  atomicAdd(o + 0, v.x);
  atomicAdd(o + 1, v.y);
  atomicAdd(o + 2, v.z);
  atomicAdd(o + 3, v.w);
}

// h = ELU( agg/deg + R )  elementwise over N*128
__global__ void __launch_bounds__(256)
finish0_kernel(const float* __restrict__ agg, const float* __restrict__ R,
               const float* __restrict__ deg, float* __restrict__ h, int total)
{
  int t = blockIdx.x * blockDim.x + threadIdx.x;
  if (t >= total) return;
  const float d = fmaxf(deg[t >> 7], 1.0f);
  const float v = agg[t] / d + R[t];
  h[t] = v > 0.0f ? v : (__expf(v) - 1.0f);
}

// Per node (one wave32 each): v = ELU(agg/deg + R) over 40 logits, then
// log_softmax via wave shuffles (lanes 0-31 hold f, lanes 0-7 also f+32).
__global__ void __launch_bounds__(256)
finish1_lsm_kernel(const float* __restrict__ agg,  // N x 48
                   const float* __restrict__ R,    // N x 48
                   const float* __restrict__ deg,
                   float* __restrict__ out,        // N x 40
                   int n)
{
  const int wave = threadIdx.x >> 5;
  const int lane = threadIdx.x & 31;
  const int i = blockIdx.x * (blockDim.x >> 5) + wave;
  if (i >= n) return;

  const float d = fmaxf(deg[i], 1.0f);
  const float* ar = agg + (size_t)i * 48;
  const float* rr = R   + (size_t)i * 48;

  float v1 = ar[lane] / d + rr[lane];
  v1 = v1 > 0.0f ? v1 : (__expf(v1) - 1.0f);

  const bool has2 = lane < 8;
  float v2 = 0.0f;
  if (has2) {
    float t = ar[32 + lane] / d + rr[32 + lane];
    v2 = t > 0.0f ? t : (__expf(t) - 1.0f);
  }

  float m = has2 ? fmaxf(v1, v2) : v1;
  #pragma unroll
  for (int off = 16; off; off >>= 1) m = fmaxf(m, __shfl_xor(m, off, 32));

  float s = __expf(v1 - m) + (has2 ? __expf(v2 - m) : 0.0f);
  #pragma unroll
  for (int off = 16; off; off >>= 1) s += __shfl_xor(s, off, 32);

  const float ls = m + __logf(s);
  float* orow = out + (size_t)i * 40;
  orow[lane] = v1 - ls;
  if (has2) orow[32 + lane] = v2 - ls;
}

// ---------------------------------------------------------------------------

extern "C" void kernel_launch(void* const* d_in, const int* in_sizes, int n_in,
                              void* d_out, int out_size, void* d_ws, size_t ws_size,
                              hipStream_t stream)
{
  const float* x   = (const float*)d_in[0];
  const int*   ei  = (const int*)  d_in[1];
  const float* Wl0 = (const float*)d_in[2];
  const float* Wr0 = (const float*)d_in[3];
  const float* b0  = (const float*)d_in[4];
  const float* Wl1 = (const float*)d_in[5];
  const float* Wr1 = (const float*)d_in[6];
  const float* b1  = (const float*)d_in[7];
  float* out = (float*)d_out;

  const int N = in_sizes[0] / 128;   // 50000 (multiple of 16)
  const int E = in_sizes[1] / 2;     // 800000
  const int* src = ei;
  const int* dst = ei + E;

  // Workspace layout (floats): N*385 + ~12.4K total (~77 MB), with reuse.
  float* bufA = (float*)d_ws;              // N*128 : T0, then h0
  float* bufB = bufA + (size_t)N * 128;    // N*128 : R0, then T1|R1 (N*48 each)
  float* bufC = bufB + (size_t)N * 128;    // N*128 : agg0, then agg1 (N*48)
  float* deg  = bufC + (size_t)N * 128;    // N     : degree
  float* Wl1p = deg  + (size_t)N;          // 128*48: padded W_l1
  float* Wr1p = Wl1p + 128 * 48;           // 128*48: padded W_r1
  float* b1p  = Wr1p + 128 * 48;           // 48    : padded b1

  // zero agg0 + deg (contiguous)
  hipMemsetAsync(bufC, 0, ((size_t)N * 128 + (size_t)N) * sizeof(float), stream);

  pad_l1_params<<<(128 * 48 + 255) / 256, 256, 0, stream>>>(
      Wl1, Wr1, b1, Wl1p, Wr1p, b1p);

  deg_kernel<<<(E + 255) / 256, 256, 0, stream>>>(dst, deg, E);

  // ---- layer 0 ----
  // T0 = x@Wl0, R0 = x@Wr0 + b0   (8 col-tiles -> 256-thread blocks)
  gemm_dual_wmma<128><<<N / 16, 256, 0, stream>>>(x, Wl0, Wr0, b0, bufA, bufB);

  {
    long long total = (long long)E * 32;
    scatter_add<32, 128><<<(unsigned)((total + 255) / 256), 256, 0, stream>>>(
        bufA, src, dst, bufC, E);
  }
  finish0_kernel<<<(N * 128 + 255) / 256, 256, 0, stream>>>(
      bufC, bufB, deg, bufA, N * 128);   // h0 -> bufA

  // ---- layer 1 ----  (C=40 padded to 48; 3 col-tiles -> 96-thread blocks)
  float* T1 = bufB;
  float* R1 = bufB + (size_t)N * 48;
  gemm_dual_wmma<48><<<N / 16, 96, 0, stream>>>(bufA, Wl1p, Wr1p, b1p, T1, R1);

  hipMemsetAsync(bufC, 0, (size_t)N * 48 * sizeof(float), stream);  // agg1
  {
    long long total = (long long)E * 12;
    scatter_add<12, 48><<<(unsigned)((total + 255) / 256), 256, 0, stream>>>(
        T1, src, dst, bufC, E);
  }

  finish1_lsm_kernel<<<(N + 7) / 8, 256, 0, stream>>>(bufC, R1, deg, out, N);
}